// SpatiotemporalAttention_32521492365736
// MI455X (gfx1250) — compile-verified
//
#include <hip/hip_runtime.h>
#include <stdint.h>

typedef __attribute__((ext_vector_type(2))) float v2f;
typedef __attribute__((ext_vector_type(8))) float v8f;

#define NODE_DIM 64
#define MSG_DIM  64
#define PROP_DIM 32
#define HIDDEN   128
#define V_N      10000
#define T_N      8
#define S_N      16
#define LDS_STRIDE 68   // floats per staged message row (padding -> conflict-free b64 reads)
#define PIPE     4      // async load pipeline depth (power of two: buf select is an AND)

// W1 row indices (row-major, 161 x 128):
//   rows [0,64) = Wh ; rows [64,128) = Wm ; row 128 = Wt ; rows [129,161) = Wp

__device__ __forceinline__ uint32_t lds_off(const void* p) {
    // LDS aperture: flat addr low 32 bits are the workgroup-relative LDS offset
    return (uint32_t)(uintptr_t)p;
}

__device__ __forceinline__ void async_copy_b128(uint32_t lds_byte, const float* g) {
    // GLOBAL_LOAD_ASYNC_TO_LDS_B128: per-lane 16B global -> LDS, tracked by ASYNCcnt
    asm volatile("global_load_async_to_lds_b128 %0, %1, off"
                 :: "v"(lds_byte), "v"(g)
                 : "memory");
}

__launch_bounds__(128, 1)
__global__ void st_attn_kernel(const float* __restrict__ h_t,
                               const float* __restrict__ messages,
                               const float* __restrict__ tau,
                               const float* __restrict__ theta,
                               const float* __restrict__ W1,
                               const float* __restrict__ b1,
                               const float* __restrict__ W2,
                               const float* __restrict__ b2,
                               float* __restrict__ out)
{
    __shared__ float sMsg[PIPE][S_N * LDS_STRIDE];  // pipelined message tiles (padded)
    __shared__ float sAdd[HIDDEN];                  // per-(v,t) bias: base[v] + tau[t]*Wt
    __shared__ float sLogit[S_N];                   // per-row reduction target

    const int tid  = threadIdx.x;                   // 0..127 ; also the h index this thread owns
    const int wave = tid >> 5;                      // wave w owns h in [32w, 32w+32)
    const int lane = tid & 31;
    const int l16  = lane & 15;
    const int hl   = lane >> 4;                     // lane half (K/M split per ISA layout)

    // ---- register-cached W2 fragment (C/D layout: VGPR j <-> M = j + 8*hl) ----
    float w2r[2][8];
#pragma unroll
    for (int m = 0; m < 2; ++m)
#pragma unroll
        for (int j = 0; j < 8; ++j)
            w2r[m][j] = W2[(wave * 2 + m) * 16 + j + 8 * hl];

    // ---- register-cached A fragments of Wm^T (A[M=h][K=d] = W1[64+d][h]) ----
    // f32 16x4 A layout: lane holds M = lane%16, K = 4*kk + 2*(lane/16) + {0,1}
    v2f afrag[2][16];
#pragma unroll
    for (int m = 0; m < 2; ++m) {
        const int col = (wave * 2 + m) * 16 + l16;
#pragma unroll
        for (int kk = 0; kk < 16; ++kk) {
            const int r = NODE_DIM + kk * 4 + 2 * hl;
            afrag[m][kk].x = W1[(r + 0) * HIDDEN + col];
            afrag[m][kk].y = W1[(r + 1) * HIDDEN + col];
        }
    }

    const float wtr = W1[(NODE_DIM + MSG_DIM) * HIDDEN + tid]; // Wt[h]
    const float b1r = b1[tid];
    const float b2v = b2[0];

    const int bid = blockIdx.x;
    const int G   = gridDim.x;
    const int nv  = (V_N - bid + G - 1) / G;   // #v values this block handles
    const int ntiles = nv * T_N;               // flattened (v,t) tile count for this block

    const int s0 = tid >> 4;                   // this thread copies rows s0 and s0+8
    const int c0 = (tid & 15) * 4;             // at column c0 (float4 granularity)

    // issue one tile's async loads (2 x b128 per thread = 4KB per block-tile)
    auto issue = [&](int k) {
        const int vv = bid + (k >> 3) * G;
        const int tt = k & 7;
        const float* g = messages + (size_t)((size_t)vv * T_N + tt) * (S_N * MSG_DIM);
        float* lb = &sMsg[k & (PIPE - 1)][0];
        async_copy_b128(lds_off(lb + (s0 * LDS_STRIDE + c0)),       g + (size_t)tid * 4);
        async_copy_b128(lds_off(lb + ((s0 + 8) * LDS_STRIDE + c0)), g + (size_t)(tid + 128) * 4);
    };

    // ---- pipeline prologue: fill PIPE-1 stages ----
    if (ntiles > 0) issue(0);
    if (ntiles > 1) issue(1);
    if (ntiles > 2) issue(2);

    float basev = 0.0f;
    int v = bid;

    for (int k = 0; k < ntiles; ++k) {
        if (k + 3 < ntiles) issue(k + 3);      // prefetch 3 tiles ahead

        const int t = k & 7;
        if (t == 0) {
            // ---- per-v base[h] = h_t[v]@Wh + theta[v]@Wp + b1 (96 FMA, negligible) ----
            v = bid + (k >> 3) * G;
            basev = b1r;
            const float* hv = h_t + (size_t)v * NODE_DIM;
            const float* tv = theta + (size_t)v * PROP_DIM;
#pragma unroll 8
            for (int d = 0; d < NODE_DIM; ++d)
                basev += hv[d] * W1[d * HIDDEN + tid];
#pragma unroll 8
            for (int p = 0; p < PROP_DIM; ++p)
                basev += tv[p] * W1[(NODE_DIM + MSG_DIM + 1 + p) * HIDDEN + tid];
        }

        sAdd[tid] = basev + tau[t] * wtr;
        if (tid < S_N) sLogit[tid] = 0.0f;

        // wait for this tile's async batch (in-order retirement; 2 instr per batch;
        // after issue there are min(PIPE, ntiles-k) batches in flight)
        const int rem = ntiles - k;
        if (rem > 3)       asm volatile("s_wait_asynccnt 0x6" ::: "memory");
        else if (rem == 3) asm volatile("s_wait_asynccnt 0x4" ::: "memory");
        else if (rem == 2) asm volatile("s_wait_asynccnt 0x2" ::: "memory");
        else               asm volatile("s_wait_asynccnt 0x0" ::: "memory");
        __syncthreads();   // B1: tile data + sAdd visible to all waves

        const float* msg = &sMsg[k & (PIPE - 1)][0];

        // ---- accumulators pre-loaded with the bias (C layout: M = j + 8*hl) ----
        v8f acc0, acc1;
#pragma unroll
        for (int j = 0; j < 8; ++j) {
            acc0[j] = sAdd[(wave * 2 + 0) * 16 + j + 8 * hl];
            acc1[j] = sAdd[(wave * 2 + 1) * 16 + j + 8 * hl];
        }

        // ---- K loop: 16 steps of V_WMMA_F32_16X16X4_F32, B frag shared by both M-tiles ----
#pragma unroll
        for (int kk = 0; kk < 16; ++kk) {
            // B layout: lane holds N = lane%16 (= s), K = 4*kk + 2*hl + {0,1}
            const v2f bf = *(const v2f*)(&msg[l16 * LDS_STRIDE + kk * 4 + 2 * hl]);
            acc0 = __builtin_amdgcn_wmma_f32_16x16x4_f32(
                false, afrag[0][kk], false, bf, (short)0, acc0, false, false);
            acc1 = __builtin_amdgcn_wmma_f32_16x16x4_f32(
                false, afrag[1][kk], false, bf, (short)0, acc1, false, false);
        }

        // ---- relu, dot with W2, reduce across h via LDS float atomics ----
        float partial = 0.0f;
#pragma unroll
        for (int j = 0; j < 8; ++j) {
            float x0 = acc0[j] > 0.0f ? acc0[j] : 0.0f;
            float x1 = acc1[j] > 0.0f ? acc1[j] : 0.0f;
            partial += x0 * w2r[0][j] + x1 * w2r[1][j];
        }
        atomicAdd(&sLogit[l16], partial);
        __syncthreads();   // B2: logits complete; also protects LDS reuse next iter

        if (tid < S_N)
            out[(size_t)((size_t)v * T_N + t) * S_N + tid] = sLogit[tid] + b2v;
    }
}

extern "C" void kernel_launch(void* const* d_in, const int* in_sizes, int n_in,
                              void* d_out, int out_size, void* d_ws, size_t ws_size,
                              hipStream_t stream) {
    (void)in_sizes; (void)n_in; (void)out_size; (void)d_ws; (void)ws_size;
    const float* h_t      = (const float*)d_in[0];
    const float* messages = (const float*)d_in[1];
    const float* tau      = (const float*)d_in[2];
    const float* theta    = (const float*)d_in[3];
    const float* W1       = (const float*)d_in[4];
    const float* b1       = (const float*)d_in[5];
    const float* W2       = (const float*)d_in[6];
    const float* b2       = (const float*)d_in[7];
    float* out            = (float*)d_out;

    const int blocks = 2048;   // grid-stride over v; ~40 tiles/block amortizes A-frag load
    st_attn_kernel<<<blocks, 128, 0, stream>>>(h_t, messages, tau, theta, W1, b1, W2, b2, out);
}